// LambaLayer_55362128445926
// MI455X (gfx1250) — compile-verified
//
#include <hip/hip_runtime.h>
#include <math.h>

// LambaLayer forward: q = BN(Wq·x), v = BN(Wv·c), k~ = softmax_w(Wk·c)
// out = concat([q, v, k~], channel axis)  -> (32, 384, 64, 64) f32
//
// Memory-bound at f32 (AI ~55 FLOP/B vs 23.3 TB/s HBM): keep exact f32 and run
// the channel-mixing GEMMs on V_WMMA_F32_16X16X4_F32.
//
// v2: activation tiles stored TRANSPOSED in LDS ([w][ch], pitch 260) so the
//     B operand (k,k+1 per lane) is one conflict-free ds_load_b64, and the
//     A (weight) operand is double-buffer prefetched in chunks of 8 K-steps.

#define B_   32
#define D_   256
#define H_   64
#define W_   64
#define HW_  (H_ * W_)
#define CQ_  64
#define CV_  256
#define CK_  64
#define OC_  (CQ_ + CV_ + CK_)   // 384
#define EPS_ 1e-5f

#define ACT_PITCH 260            // 260 % 64 == 4 -> banks 4*col+k, conflict-free
#define KS_PITCH  65

typedef __attribute__((ext_vector_type(2))) float v2f;
typedef __attribute__((ext_vector_type(8))) float v8f;

// Shared memory layout (floats):
#define XS_OFF   0                               // xT tile [w][ch]  64*260
#define CS_OFF   (XS_OFF + W_ * ACT_PITCH)       // cT tile [w][ch]  64*260
#define KS_OFF   (CS_OFF + W_ * ACT_PITCH)       // raw k scores 64*65
#define SCQ_OFF  (KS_OFF + CK_ * KS_PITCH)
#define SHQ_OFF  (SCQ_OFF + CQ_)
#define SCV_OFF  (SHQ_OFF + CQ_)
#define SHV_OFF  (SCV_OFF + CV_)
#define INV_OFF  (SHV_OFF + CV_)
#define SMEM_FLTS (INV_OFF + CK_)                // ~152.6 KB -> 2 WGs per WGP

__global__ void lamba_fused_kernel(
    const float* __restrict__ x,  const float* __restrict__ c,
    const float* __restrict__ Wq, const float* __restrict__ Wv,
    const float* __restrict__ Wk,
    const float* __restrict__ qg, const float* __restrict__ qb,
    const float* __restrict__ qm, const float* __restrict__ qv,
    const float* __restrict__ vg, const float* __restrict__ vb,
    const float* __restrict__ vm, const float* __restrict__ vv,
    float* __restrict__ out)
{
    extern __shared__ float smem[];
    float* xsT = smem + XS_OFF;
    float* csT = smem + CS_OFF;
    float* ks  = smem + KS_OFF;
    float* scq = smem + SCQ_OFF;
    float* shq = smem + SHQ_OFF;
    float* scv = smem + SCV_OFF;
    float* shv = smem + SHV_OFF;
    float* inv_s = smem + INV_OFF;

    const int tid  = threadIdx.x;
    const int wave = tid >> 5;          // 0..7 (wave32)
    const int lane = tid & 31;
    const int lr   = lane & 15;         // lane within half
    const int half = lane >> 4;         // 0 or 1

    const int bh = blockIdx.x;          // b*H + h
    const int b  = bh >> 6;
    const int h  = bh & 63;

    // ---- BN scale/shift precompute (before any WMMA; EXEC restored after) ----
    if (tid < CQ_) {
        float s = qg[tid] * rsqrtf(qv[tid] + EPS_);
        scq[tid] = s;
        shq[tid] = qb[tid] - qm[tid] * s;
    }
    if (tid < CV_) {
        float s = vg[tid] * rsqrtf(vv[tid] + EPS_);
        scv[tid] = s;
        shv[tid] = vb[tid] - vm[tid] * s;
    }

    // ---- Cooperative tile loads: coalesced global b128, transposed LDS scatter ----
    {
        const size_t base = (size_t)b * D_ * HW_ + (size_t)h * W_;
        const float4* xg = reinterpret_cast<const float4*>(x + base);
        const float4* cg = reinterpret_cast<const float4*>(c + base);
        #pragma unroll
        for (int i = tid; i < D_ * W_ / 4; i += 256) {
            int ch = i >> 4;            // 16 float4 per channel row
            int w4 = (i & 15) * 4;
            float4 vx = xg[(size_t)ch * (HW_ / 4) + (i & 15)];
            float4 vc = cg[(size_t)ch * (HW_ / 4) + (i & 15)];
            xsT[(w4 + 0) * ACT_PITCH + ch] = vx.x;
            xsT[(w4 + 1) * ACT_PITCH + ch] = vx.y;
            xsT[(w4 + 2) * ACT_PITCH + ch] = vx.z;
            xsT[(w4 + 3) * ACT_PITCH + ch] = vx.w;
            csT[(w4 + 0) * ACT_PITCH + ch] = vc.x;
            csT[(w4 + 1) * ACT_PITCH + ch] = vc.y;
            csT[(w4 + 2) * ACT_PITCH + ch] = vc.z;
            csT[(w4 + 3) * ACT_PITCH + ch] = vc.w;
        }
    }
    __syncthreads();

    const size_t obase = (size_t)b * OC_ * HW_ + (size_t)h * W_;

    // ---- 24 strips of (16 out-ch x 64 cols): q[0..3], v[4..19], k[20..23] ----
    for (int s = wave; s < 24; s += 8) {
        const float* Wp;
        const float* act;
        int orow, kind;                  // kind: 0=q 1=v 2=k
        if (s < 4)       { Wp = Wq; act = xsT; orow = s * 16;        kind = 0; }
        else if (s < 20) { Wp = Wv; act = csT; orow = (s - 4) * 16;  kind = 1; }
        else             { Wp = Wk; act = csT; orow = (s - 20) * 16; kind = 2; }

        v8f acc[4] = {v8f{}, v8f{}, v8f{}, v8f{}};

        // A operand: lane (half,lr) holds W[orow+lr][k0+2*half, k0+2*half+1]
        const float* wrow  = Wp + (size_t)(orow + lr) * D_ + 2 * half;
        // B operand base: element (col=t*16+lr, k=k0+2*half); (k,k+1) contiguous
        const float* bbase = act + lr * ACT_PITCH + 2 * half;

        // Double-buffered A prefetch: chunks of 8 K-steps (K += 32 per chunk)
        v2f abuf[2][8];
        #pragma unroll
        for (int i = 0; i < 8; ++i)
            abuf[0][i] = *(const v2f*)(wrow + i * 4);

        for (int chk = 0; chk < 8; ++chk) {
            if (chk < 7) {
                #pragma unroll
                for (int i = 0; i < 8; ++i)
                    abuf[(chk + 1) & 1][i] = *(const v2f*)(wrow + (chk + 1) * 32 + i * 4);
            }
            #pragma unroll
            for (int i = 0; i < 8; ++i) {
                const int k0 = chk * 32 + i * 4;
                v2f a = abuf[chk & 1][i];
                #pragma unroll
                for (int t = 0; t < 4; ++t) {
                    v2f bb = *(const v2f*)(bbase + t * 16 * ACT_PITCH + k0);
                    acc[t] = __builtin_amdgcn_wmma_f32_16x16x4_f32(
                        false, a, false, bb, (short)0, acc[t], false, false);
                }
            }
        }

        // C/D layout: VGPR j -> rows orow+j (lanes 0-15) / orow+j+8 (lanes 16-31)
        if (kind == 2) {
            // stage raw k scores to LDS for row softmax
            #pragma unroll
            for (int j = 0; j < 8; ++j) {
                int r = orow + j + 8 * half;
                #pragma unroll
                for (int t = 0; t < 4; ++t)
                    ks[r * KS_PITCH + t * 16 + lr] = acc[t][j];
            }
        } else {
            const float* sc = (kind == 1) ? scv : scq;
            const float* sh = (kind == 1) ? shv : shq;
            const int choff = (kind == 1) ? CQ_ : 0;
            #pragma unroll
            for (int j = 0; j < 8; ++j) {
                int o = orow + j + 8 * half;
                float scale = sc[o], shift = sh[o];
                float* op = out + obase + (size_t)(choff + o) * HW_;
                #pragma unroll
                for (int t = 0; t < 4; ++t)
                    op[t * 16 + lr] = acc[t][j] * scale + shift;
            }
        }
    }
    __syncthreads();

    // ---- softmax over w for each of the 64 k rows ----
    if (tid < CK_) {
        float* row = ks + tid * KS_PITCH;
        float mx = -INFINITY;
        #pragma unroll 8
        for (int w = 0; w < W_; ++w) mx = fmaxf(mx, row[w]);
        float sum = 0.f;
        #pragma unroll 8
        for (int w = 0; w < W_; ++w) {
            float e = __expf(row[w] - mx);
            row[w] = e;
            sum += e;
        }
        inv_s[tid] = 1.f / sum;
    }
    __syncthreads();

    // coalesced k~ store: 4096 elements by 256 threads
    for (int i = tid; i < CK_ * W_; i += 256) {
        int r = i >> 6, w = i & 63;
        out[obase + (size_t)(CQ_ + CV_ + r) * HW_ + w] = ks[r * KS_PITCH + w] * inv_s[r];
    }
}

extern "C" void kernel_launch(void* const* d_in, const int* in_sizes, int n_in,
                              void* d_out, int out_size, void* d_ws, size_t ws_size,
                              hipStream_t stream) {
    const float* x  = (const float*)d_in[0];
    const float* c  = (const float*)d_in[1];
    const float* Wq = (const float*)d_in[2];
    const float* Wv = (const float*)d_in[3];
    const float* Wk = (const float*)d_in[4];
    const float* qg = (const float*)d_in[5];
    const float* qb = (const float*)d_in[6];
    const float* qm = (const float*)d_in[7];
    const float* qv = (const float*)d_in[8];
    const float* vg = (const float*)d_in[9];
    const float* vb = (const float*)d_in[10];
    const float* vm = (const float*)d_in[11];
    const float* vv = (const float*)d_in[12];
    float* out = (float*)d_out;

    dim3 grid(B_ * H_);       // 2048 workgroups: one (b, h) row each
    dim3 block(256);          // 8 wave32
    size_t shmem = (size_t)SMEM_FLTS * sizeof(float);   // ~152.6 KB

    lamba_fused_kernel<<<grid, block, shmem, stream>>>(
        x, c, Wq, Wv, Wk, qg, qb, qm, qv, vg, vb, vm, vv, out);
}